// RoIPooling_52424370815307
// MI455X (gfx1250) — compile-verified
//
#include <hip/hip_runtime.h>
#include <hip/hip_bf16.h>

#define FEAT_HW   7
#define IMG_SIZE  224
#define N_LM      8
#define C_TOTAL   512
#define C_CHUNK   128
#define HW2       49                       // 7*7
#define LDS_FLOATS (C_CHUNK * HW2)         // 6272 floats = 25088 B
#define LDS_VEC4   (LDS_FLOATS / 4)        // 1568 x 16B transfers
#define THREADS   256
#define FULL_ITERS (LDS_VEC4 / THREADS)    // 6 unconditional rounds
#define TAIL       (LDS_VEC4 - FULL_ITERS * THREADS)  // 32 leftover lanes

typedef int v4i __attribute__((ext_vector_type(4)));
typedef __attribute__((address_space(1))) v4i g_v4i;   // global (AS1) int4
typedef __attribute__((address_space(3))) v4i l_v4i;   // LDS    (AS3) int4

__device__ __forceinline__ int crop_start(int v) {
    int s = (v * FEAT_HW) / IMG_SIZE - 1;  // v in [0,224): floor div == C div
    s = s < 0 ? 0 : s;
    s = s > (FEAT_HW - 2) ? (FEAT_HW - 2) : s;
    return s;
}

__device__ __forceinline__ void async_cp16(const float* src, float* dst_lds) {
#if __has_builtin(__builtin_amdgcn_global_load_async_to_lds_b128)
    __builtin_amdgcn_global_load_async_to_lds_b128(
        (g_v4i*)src, (l_v4i*)dst_lds, /*imm offset*/0, /*cpol*/0);
#else
    *(float4*)dst_lds = *(const float4*)src;
#endif
}

__global__ __launch_bounds__(THREADS)
void roipool_async_kernel(const float* __restrict__ features,
                          const int*   __restrict__ landmarks,
                          float*       __restrict__ out)
{
    __shared__ float sfeat[LDS_FLOATS];
    __shared__ int   s_r0[N_LM];
    __shared__ int   s_c0[N_LM];

    const int b     = blockIdx.x;          // batch
    const int cbase = blockIdx.y * C_CHUNK;
    const int tid   = threadIdx.x;

    const float* src = features + ((size_t)b * C_TOTAL + cbase) * HW2;

    // ---- Stage 1: async DMA the contiguous 25KB chunk to LDS ----
    // 6 straight-line issues per thread (no exec-mask loop), then a 32-lane tail.
#pragma unroll
    for (int k = 0; k < FULL_ITERS; ++k) {
        const int i = k * THREADS + tid;
        async_cp16(src + i * 4, &sfeat[i * 4]);
    }
    if (tid < TAIL) {
        const int i = FULL_ITERS * THREADS + tid;
        async_cp16(src + i * 4, &sfeat[i * 4]);
    }

    // ---- Stage 2: 8 threads compute clipped window starts ----
    if (tid < N_LM) {
        int xv = landmarks[b * (2 * N_LM) + 2 * tid];      // rows come from x
        int yv = landmarks[b * (2 * N_LM) + 2 * tid + 1];  // cols come from y
        s_r0[tid] = crop_start(xv);
        s_c0[tid] = crop_start(yv);
    }

    // Drain this wave's async transfers, then sync all 8 waves.
#if __has_builtin(__builtin_amdgcn_s_wait_asynccnt)
    __builtin_amdgcn_s_wait_asynccnt(0);
#else
    asm volatile("s_wait_asynccnt 0" ::: "memory");
#endif
    __syncthreads();

    // ---- Stage 3: 2x2 max-pool per (landmark, channel), coalesced NT stores ----
    float* obase = out + (size_t)b * (N_LM * C_TOTAL) + cbase;
#pragma unroll
    for (int j = 0; j < (N_LM * C_CHUNK) / THREADS; ++j) {   // 4 outputs/thread
        int idx = j * THREADS + tid;                         // 0..1023
        int l   = idx >> 7;                                  // landmark (wave-uniform)
        int ch  = idx & (C_CHUNK - 1);
        int r0  = s_r0[l];
        int c0  = s_c0[l];
        const float* p = sfeat + ch * HW2 + r0 * FEAT_HW + c0; // stride 49: bank-conflict-free
        float m0 = fmaxf(p[0],        p[1]);
        float m1 = fmaxf(p[FEAT_HW],  p[FEAT_HW + 1]);
        __builtin_nontemporal_store(fmaxf(m0, m1),
                                    &obase[(size_t)l * C_TOTAL + ch]);
    }
}

extern "C" void kernel_launch(void* const* d_in, const int* in_sizes, int n_in,
                              void* d_out, int out_size, void* d_ws, size_t ws_size,
                              hipStream_t stream) {
    const float* features  = (const float*)d_in[0];
    const int*   landmarks = (const int*)d_in[1];
    float*       out       = (float*)d_out;

    const int B = in_sizes[1] / (2 * N_LM);          // 2048
    dim3 grid(B, C_TOTAL / C_CHUNK);                 // (2048, 4)
    roipool_async_kernel<<<grid, THREADS, 0, stream>>>(features, landmarks, out);
}